// Interpolate_86775519248465
// MI455X (gfx1250) — compile-verified
//
#include <hip/hip_runtime.h>

// Bilinear grid-sample, MI455X (gfx1250).
// Pure HBM-bandwidth gather (~0.5 GFLOP vs >0.5 GB moved -> AI < 1 FLOP/B).
// CDNA5 paths used: per-wave async global->LDS DMA for coordinates
// (ASYNCcnt is per-wave, so no block barrier needed), coalesced b128 corner
// gathers, non-temporal b128 stores so the streamed 268 MB output does not
// evict the 268 MB img working set from the 192 MB L2.
// WMMA is structurally inapplicable: every pixel has its own A (weights) and
// B (4x32 corners) operands; nothing is shared wave-wide.

typedef float v4f __attribute__((ext_vector_type(4)));
typedef float v2f __attribute__((ext_vector_type(2)));

// B=8, H=512, W=512, C=32
#define NPIX         (8 * 512 * 512)   // 2,097,152 pixels
#define PX_PER_BLOCK 32                // 256 threads / 8 lanes-per-pixel

__global__ __launch_bounds__(256) void Interpolate_86775519248465_kernel(
    const float* __restrict__ img,
    const float* __restrict__ grid,
    float* __restrict__ out)
{
    __shared__ v2f sgrid[PX_PER_BLOCK];

    const int pblock = blockIdx.x << 5;            // first pixel of this block
    const unsigned wave = threadIdx.x >> 5;        // 8 wave32s per block
    const unsigned lane = threadIdx.x & 31u;

    // ---- Each wave DMAs its own 4 pixels' grid coords (32 B) into LDS via
    // ---- the CDNA5 async path; lanes 0..3 issue one 8B transfer each.
    if (lane < 4u) {
        unsigned pix    = (wave << 2) + lane;                         // 0..31
        unsigned ldsoff = (unsigned)(size_t)(&sgrid[0]) + pix * 8u;   // LDS byte addr
        unsigned goff   = (unsigned)(pblock + (int)pix) * 8u;         // global byte off
        unsigned long long gbase = (unsigned long long)(size_t)grid;
        asm volatile("global_load_async_to_lds_b64 %0, %1, %2"
                     :: "v"(ldsoff), "v"(goff), "s"(gbase)
                     : "memory");
    }
    // ASYNCcnt is per-wave: wait on our own transfers only, no block barrier.
    asm volatile("s_wait_asynccnt 0" ::: "memory");

    const int p  = pblock + (int)(threadIdx.x >> 3);   // pixel for this thread
    const int cg = (int)(threadIdx.x & 7u) << 2;       // channel group (4 floats)

    // Coordinates from LDS (ds_load_b64), written by our own wave's DMA.
    v2f g = sgrid[threadIdx.x >> 3];

    // Unnormalize exactly as the reference does.
    float x = 0.5f * ((g.x + 1.0f) * 511.0f);
    float y = 0.5f * ((g.y + 1.0f) * 511.0f);

    float xf = floorf(x);
    float yf = floorf(y);
    int x0 = (int)xf;
    int y0 = (int)yf;
    int x0c = min(max(x0, 0), 511);
    int x1c = min(x0 + 1, 511);
    int y0c = min(max(y0, 0), 511);
    int y1c = min(y0 + 1, 511);

    float x0f = (float)x0c, x1f = (float)x1c;
    float y0f = (float)y0c, y1f = (float)y1c;

    float wx1 = x1f - x, wx0 = x - x0f;
    float wy1 = y1f - y, wy0 = y - y0f;
    float wa = wx1 * wy1;   // (y0,x0)
    float wb = wx1 * wy0;   // (y1,x0)
    float wc = wx0 * wy1;   // (y0,x1)
    float wd = wx0 * wy0;   // (y1,x1)

    // img[b, y, x, c]; b = p >> 18, batch stride = 512*512*32 = 2^23 elements.
    const float* ib = img + ((size_t)(p >> 18) << 23);
    const int r0 = y0c << 14;         // y * W * C = y * 512 * 32
    const int r1 = y1c << 14;
    const int c0 = (x0c << 5) + cg;   // x * C + channel group
    const int c1 = (x1c << 5) + cg;

    // Four coalesced 128B-per-pixel corner gathers (global_load_b128, RT:
    // these want L2 residency).
    v4f Ia = *(const v4f*)(ib + r0 + c0);
    v4f Ib = *(const v4f*)(ib + r1 + c0);
    v4f Ic = *(const v4f*)(ib + r0 + c1);
    v4f Id = *(const v4f*)(ib + r1 + c1);

    v4f o = Ia * wa;
    o += Ib * wb;
    o += Ic * wc;
    o += Id * wd;

    // Streamed output: non-temporal b128 store, keep L2 for img gathers.
    float* op = out + ((size_t)p << 5) + cg;
    __builtin_nontemporal_store(o, (v4f*)op);
}

extern "C" void kernel_launch(void* const* d_in, const int* in_sizes, int n_in,
                              void* d_out, int out_size, void* d_ws, size_t ws_size,
                              hipStream_t stream) {
    (void)in_sizes; (void)n_in; (void)out_size; (void)d_ws; (void)ws_size;
    const float* img  = (const float*)d_in[0];
    const float* grid = (const float*)d_in[1];
    float* out        = (float*)d_out;

    // 2,097,152 pixels * 8 lanes/pixel = 16,777,216 threads = 65,536 blocks x 256.
    dim3 grd(NPIX / PX_PER_BLOCK);
    dim3 blk(256);
    Interpolate_86775519248465_kernel<<<grd, blk, 0, stream>>>(img, grid, out);
}